// LinearCRF_14379550507074
// MI455X (gfx1250) — compile-verified
//
#include <hip/hip_runtime.h>
#include <hip/hip_bf16.h>
#include <stdint.h>

// Linear-chain CRF forward/backward (L = 4M, 2 labels) as a parallel scan over
// 2x2 log-semiring matrices.  Memory-bound (~200MB total traffic ~ 9us @ 23.3TB/s;
// feats fits in the 192MB L2 so re-reads are near-free).
// CDNA5 usage: wave32 blocks, LDS tile staging via global_load_async_to_lds_b128,
// output drain via global_store_async_from_lds_b128, s_wait_asynccnt,
// ds_load_b64 compute reads, v_exp/v_log transcendentals for LSE.

#define LENGTH      4194304
#define SEG         64                 // elements per thread (contiguous)
#define BLOCK       128                // threads per block
#define TILE_ELEMS  (BLOCK * SEG)      // 8192 elements (64KB raw) per block tile
#define NBLK        (LENGTH / TILE_ELEMS) // 512
#define GT          (LENGTH / SEG)     // 65536 total scan threads
#define PSTR        66                 // padded tile stride in float2 (528B, 16B aligned)
#define K2T         1024               // scan-kernel threads
#define RUN         (GT / K2T)         // 64 block-matrices per scan thread
#define BIGF        1e30f

// ---------------------------------------------------------------- primitives
__device__ __forceinline__ float lse2(float a, float b) {
    float m = fmaxf(a, b);
    float d = fabsf(a - b);
    return m + __logf(1.0f + __expf(-d));   // v_exp_f32 + v_log_f32
}

struct Mat { float m00, m01, m10, m11; };

__device__ __forceinline__ Mat mident() {
    Mat r; r.m00 = 0.f; r.m01 = -BIGF; r.m10 = -BIGF; r.m11 = 0.f; return r;
}
// log-semiring product (A then B): r[p][n] = LSE_k(A[p][k] + B[k][n])
__device__ __forceinline__ Mat mcomp(const Mat A, const Mat B) {
    Mat r;
    r.m00 = lse2(A.m00 + B.m00, A.m01 + B.m10);
    r.m01 = lse2(A.m00 + B.m01, A.m01 + B.m11);
    r.m10 = lse2(A.m10 + B.m00, A.m11 + B.m10);
    r.m11 = lse2(A.m10 + B.m01, A.m11 + B.m11);
    return r;
}
__device__ __forceinline__ Mat ldm(const float4 v) {
    Mat m; m.m00 = v.x; m.m01 = v.y; m.m10 = v.z; m.m11 = v.w; return m;
}

// ------------------------------------------------- CDNA5 async global<->LDS DMA
__device__ __forceinline__ void async_load_b128(uint32_t lds_addr, uint32_t voff,
                                                const void* sbase) {
    asm volatile("global_load_async_to_lds_b128 %0, %1, %2"
                 :: "v"(lds_addr), "v"(voff), "s"(sbase)
                 : "memory");
}
__device__ __forceinline__ void async_store_b128(uint32_t voff, uint32_t lds_addr,
                                                 const void* sbase) {
    asm volatile("global_store_async_from_lds_b128 %0, %1, %2"
                 :: "v"(voff), "v"(lds_addr), "s"(sbase)
                 : "memory");
}
__device__ __forceinline__ void wait_async0() {
    asm volatile("s_wait_asynccnt 0" ::: "memory");
}

// Stage TILE_ELEMS contiguous float2 elements into padded LDS tile, coalesced.
// pair = 2 elements = 16B; 4096 pairs, 128 lanes -> 32 b128 DMAs per lane.
__device__ __forceinline__ void stage_tile_async(float2* tile, const float2* gbase,
                                                 int lid) {
    #pragma unroll 4
    for (int it = 0; it < 32; ++it) {
        int pairIdx = it * BLOCK + lid;
        int s  = pairIdx >> 5;            // segment (32 pairs per 64-elem segment)
        int k2 = (pairIdx & 31) * 2;      // element offset inside segment (even)
        uint32_t lds_addr = (uint32_t)(size_t)(tile + s * PSTR + k2);
        uint32_t voff     = (uint32_t)(pairIdx * 16);
        async_load_b128(lds_addr, voff, (const void*)gbase);
    }
}

// Coalesced async drain of a padded LDS tile to global (LDS->mem DMA, no VGPRs).
// ASYNCcnt is drained by the implicit wait-idle at s_endpgm.
__device__ __forceinline__ void store_tile_async(const float2* tile, float2* gbase,
                                                 int lid) {
    #pragma unroll 4
    for (int it = 0; it < 32; ++it) {
        int pairIdx = it * BLOCK + lid;
        int s  = pairIdx >> 5;
        int k2 = (pairIdx & 31) * 2;
        uint32_t lds_addr = (uint32_t)(size_t)(tile + s * PSTR + k2);
        async_store_b128((uint32_t)(pairIdx * 16), lds_addr, (const void*)gbase);
    }
}

// =====================================================================
// Pass 1: each thread composes its 64-element segment into forward matrix
//   F_t = M_s (x) ... (x) M_{e-1},  M_i[p][n] = T[n][p] + feats[i][n]
// and backward matrix
//   B_t = N_s (x) ... (x) N_{e-1},  N_i[p][n] = feats[i][p] + T[n][p]
// with rank-1 boundary matrices at i==0 (fwd) and i==L-1 (bwd).
// =====================================================================
__global__ __launch_bounds__(BLOCK)
void crf_pass1(const float2* __restrict__ feats, const float* __restrict__ trans,
               float4* __restrict__ Fm, float4* __restrict__ Bm) {
    extern __shared__ float2 tile[];
    const int lid = threadIdx.x;
    const int blk = blockIdx.x;
    stage_tile_async(tile, feats + (size_t)blk * TILE_ELEMS, lid);
    wait_async0();
    __syncthreads();

    const float T00 = trans[0], T01 = trans[1], T10 = trans[2], T11 = trans[3];
    const int t = blk * BLOCK + lid;   // global segment id
    float F00, F01, F10, F11, B00, B01, B10, B11;

    for (int k = 0; k < SEG; ++k) {
        float2 f = tile[lid * PSTR + k];
        int i = t * SEG + k;
        if (k == 0) {
            if (i == 0) { // rank-1: M_0[p][n] = feats[0][n]
                F00 = f.x; F01 = f.y; F10 = f.x; F11 = f.y;
            } else {      // M_i[p][n] = T[n][p] + f[n]
                F00 = T00 + f.x; F01 = T10 + f.y;
                F10 = T01 + f.x; F11 = T11 + f.y;
            }
            // N_i[p][n] = f[p] + T[n][p]   (i < L-1 always holds at k==0)
            B00 = f.x + T00; B01 = f.x + T10;
            B10 = f.y + T01; B11 = f.y + T11;
        } else {
            // F = F (x) M_i
            float a0 = lse2(F00 + T00, F01 + T01) + f.x;
            float a1 = lse2(F00 + T10, F01 + T11) + f.y;
            float b0 = lse2(F10 + T00, F11 + T01) + f.x;
            float b1 = lse2(F10 + T10, F11 + T11) + f.y;
            F00 = a0; F01 = a1; F10 = b0; F11 = b1;
            // B = B (x) N_i
            if (i == LENGTH - 1) {      // rank-1: N_{L-1}[k][n] = f[k]
                float r0 = lse2(B00 + f.x, B01 + f.y);
                float r1 = lse2(B10 + f.x, B11 + f.y);
                B00 = r0; B01 = r0; B10 = r1; B11 = r1;
            } else {
                float c0 = lse2(B00 + f.x + T00, B01 + f.y + T01);
                float c1 = lse2(B00 + f.x + T10, B01 + f.y + T11);
                float d0 = lse2(B10 + f.x + T00, B11 + f.y + T01);
                float d1 = lse2(B10 + f.x + T10, B11 + f.y + T11);
                B00 = c0; B01 = c1; B10 = d0; B11 = d1;
            }
        }
    }
    Fm[t] = make_float4(F00, F01, F10, F11);
    Bm[t] = make_float4(B00, B01, B10, B11);
}

// =====================================================================
// Pass 2: one 1024-thread block scans the 65536 segment matrices.
// Forward: exclusive prefixes -> alpha entry row R_t = row0(prefix).
// Backward: exclusive suffixes -> beta entry col S_t = col0(suffix).
// Also computes Z = LSE(row0(total product)).
// =====================================================================
__global__ __launch_bounds__(K2T)
void crf_scan(const float4* __restrict__ Fm, const float4* __restrict__ Bm,
              float2* __restrict__ Rrow, float2* __restrict__ Scol,
              float* __restrict__ Zp) {
    __shared__ Mat S[K2T];
    const int j = threadIdx.x;

    // ---------------- forward ----------------
    Mat C = ldm(Fm[j * RUN]);
    for (int k = 1; k < RUN; ++k) C = mcomp(C, ldm(Fm[j * RUN + k]));
    S[j] = C;
    __syncthreads();
    for (int d = 1; d < K2T; d <<= 1) {
        Mat v = S[j];
        Mat l = (j >= d) ? S[j - d] : mident();
        __syncthreads();
        if (j >= d) S[j] = mcomp(l, v);
        __syncthreads();
    }
    Mat P = (j == 0) ? mident() : S[j - 1];     // exclusive prefix for run j
    for (int k = 0; k < RUN; ++k) {
        int tt = j * RUN + k;
        Rrow[tt] = make_float2(P.m00, P.m01);   // seed row (0,-inf) selects row 0
        P = mcomp(P, ldm(Fm[tt]));
    }
    if (j == K2T - 1) *Zp = lse2(P.m00, P.m01); // Z from total product
    __syncthreads();

    // ---------------- backward ----------------
    Mat Cb = ldm(Bm[j * RUN]);
    for (int k = 1; k < RUN; ++k) Cb = mcomp(Cb, ldm(Bm[j * RUN + k]));
    S[j] = Cb;
    __syncthreads();
    for (int d = 1; d < K2T; d <<= 1) {
        Mat v = S[j];
        Mat r = (j + d < K2T) ? S[j + d] : mident();
        __syncthreads();
        if (j + d < K2T) S[j] = mcomp(v, r);
        __syncthreads();
    }
    Mat Q = (j == K2T - 1) ? mident() : S[j + 1]; // exclusive suffix for run j
    for (int k = RUN - 1; k >= 0; --k) {
        int tt = j * RUN + k;
        Scol[tt] = make_float2(Q.m00, Q.m10);   // seed col (0,-inf) selects col 0
        Q = mcomp(ldm(Bm[tt]), Q);
    }
}

// =====================================================================
// Pass 3a: replay forward scan per segment, write alpha to d_out
// (alpha tile drained LDS->global via async store DMA).
// =====================================================================
__global__ __launch_bounds__(BLOCK)
void crf_alpha(const float2* __restrict__ feats, const float* __restrict__ trans,
               const float2* __restrict__ Rrow, float2* __restrict__ out) {
    extern __shared__ float2 tile[];
    const int lid = threadIdx.x;
    const int blk = blockIdx.x;
    stage_tile_async(tile, feats + (size_t)blk * TILE_ELEMS, lid);
    wait_async0();
    __syncthreads();

    const float T00 = trans[0], T01 = trans[1], T10 = trans[2], T11 = trans[3];
    const int t = blk * BLOCK + lid;
    float2 r = Rrow[t];
    for (int k = 0; k < SEG; ++k) {
        float2 f = tile[lid * PSTR + k];
        int i = t * SEG + k;
        float2 a;
        if (i == 0) {                       // alpha_0 = feats[0]  (rank-1 M_0)
            float s = lse2(r.x, r.y);
            a.x = f.x + s; a.y = f.y + s;
        } else {
            a.x = f.x + lse2(r.x + T00, r.y + T01);
            a.y = f.y + lse2(r.x + T10, r.y + T11);
        }
        r = a;
        tile[lid * PSTR + k] = r;           // overwrite tile with alpha
    }
    __syncthreads();                        // drain DS writes before DMA-out
    store_tile_async(tile, out + (size_t)blk * TILE_ELEMS, lid);
}

// =====================================================================
// Pass 3b: replay backward scan, combine with alpha (async-loaded back from
// d_out), write marginal = exp(alpha + beta - feats - Z) via async store DMA.
// =====================================================================
__global__ __launch_bounds__(BLOCK)
void crf_beta(const float2* __restrict__ feats, const float* __restrict__ trans,
              const float2* __restrict__ Scol, const float* __restrict__ Zp,
              float2* __restrict__ out) {
    extern __shared__ float2 smem[];
    float2* tileF = smem;                   // feats
    float2* tileA = smem + BLOCK * PSTR;    // alpha -> marginal (in place)
    const int lid = threadIdx.x;
    const int blk = blockIdx.x;
    stage_tile_async(tileF, feats + (size_t)blk * TILE_ELEMS, lid);
    stage_tile_async(tileA, out   + (size_t)blk * TILE_ELEMS, lid);
    wait_async0();
    __syncthreads();

    const float T00 = trans[0], T01 = trans[1], T10 = trans[2], T11 = trans[3];
    const float Z = *Zp;
    const int t = blk * BLOCK + lid;
    float2 c = Scol[t];
    for (int k = SEG - 1; k >= 0; --k) {
        float2 f = tileF[lid * PSTR + k];
        int i = t * SEG + k;
        float2 b;
        if (i == LENGTH - 1) {              // beta_{L-1} = feats[L-1]
            float s = lse2(c.x, c.y);
            b.x = f.x + s; b.y = f.y + s;
        } else {
            // beta[p] = f[p] + LSE_n(T[n][p] + c[n])
            b.x = f.x + lse2(T00 + c.x, T10 + c.y);
            b.y = f.y + lse2(T01 + c.x, T11 + c.y);
        }
        c = b;
        float2 a = tileA[lid * PSTR + k];
        float2 m;
        m.x = __expf(a.x + b.x - f.x - Z);
        m.y = __expf(a.y + b.y - f.y - Z);
        tileA[lid * PSTR + k] = m;
    }
    __syncthreads();                        // drain DS writes before DMA-out
    store_tile_async(tileA, out + (size_t)blk * TILE_ELEMS, lid);
}

// =====================================================================
extern "C" void kernel_launch(void* const* d_in, const int* in_sizes, int n_in,
                              void* d_out, int out_size, void* d_ws, size_t ws_size,
                              hipStream_t stream) {
    const float2* feats = (const float2*)d_in[0];   // [L,2] f32
    const float*  trans = (const float*)d_in[1];    // [2,2] f32
    float2* out = (float2*)d_out;                   // [L,2] f32

    // workspace layout (floats): F[GT*4] | B[GT*4] | R[GT*2] | S[GT*2] | Z
    float4* Fm = (float4*)d_ws;
    float4* Bm = Fm + GT;
    float2* Rr = (float2*)(Bm + GT);
    float2* Sc = Rr + GT;
    float*  Zp = (float*)(Sc + GT);

    const size_t smem1 = (size_t)BLOCK * PSTR * sizeof(float2);  // 67,584 B

    crf_pass1<<<NBLK, BLOCK, smem1,     stream>>>(feats, trans, Fm, Bm);
    crf_scan <<<1,    K2T,   0,         stream>>>(Fm, Bm, Rr, Sc, Zp);
    crf_alpha<<<NBLK, BLOCK, smem1,     stream>>>(feats, trans, Rr, out);
    crf_beta <<<NBLK, BLOCK, 2 * smem1, stream>>>(feats, trans, Sc, Zp, out);
}